// NormalizationLayer_61409442398476
// MI455X (gfx1250) — compile-verified
//
#include <hip/hip_runtime.h>
#include <float.h>

typedef __attribute__((ext_vector_type(2))) float v2f;
typedef __attribute__((ext_vector_type(8))) float v8f;

#define PLANES  512        // B*C = 16*32
#define H1      256
#define W1      256
#define H2      64
#define W2      64
#define OUTN    319        // H1 + H2 - 1
#define TILE    64
#define RF      127        // TILE + (H2-1): receptive field of a 64-tile
#define SATN    128        // RF + 1 (zero row/col for the integral image)
#define SSTRIDE 129        // pad stride: (r*129+c)%64 varies in r and c -> no bank conflicts
#define EPSF    1.1920928955078125e-07f   // np.finfo(float32).eps

// ---------------------------------------------------------------------------
// Kernel A: per-plane template stats (mean, SSD) of feat_2, one wave32/plane.
// Reduction runs through V_WMMA_F32_16X16X4_F32 with an all-ones B matrix:
// D = A*ones + D accumulates row sums of each 16x4 chunk of the plane.
// ---------------------------------------------------------------------------
__global__ __launch_bounds__(32)
void f2_stats_kernel(const float* __restrict__ feat2, float* __restrict__ stats)
{
    const int p = blockIdx.x;          // plane
    const int l = threadIdx.x;         // lane 0..31
    const float* base = feat2 + (size_t)p * (H2 * W2);

    // 16x4 f32 A-matrix layout (ISA 7.12.2): lanes 0-15 hold row m, K=0/1 in
    // v0/v1; lanes 16-31 hold row m, K=2/3.  For a row-major 16x4 chunk of 64
    // consecutive floats, lane m reads the float2 at 4*m, lane 16+m at 4*m+2.
    const int off = (l < 16) ? (4 * l) : (4 * (l - 16) + 2);

    v8f acc  = {};
    v8f acc2 = {};
    v2f bones; bones.x = 1.0f; bones.y = 1.0f;

    for (int chunk = 0; chunk < (H2 * W2) / 64; ++chunk) {
        const float* cp = base + chunk * 64 + off;
        v2f a;  a.x  = cp[0];       a.y  = cp[1];
        v2f a2; a2.x = a.x * a.x;   a2.y = a.y * a.y;
        // 8 args: (neg_a, A, neg_b, B, c_mod, C, reuse_a, reuse_b)
        acc  = __builtin_amdgcn_wmma_f32_16x16x4_f32(false, a,  false, bones,
                                                     (short)0, acc,  false, false);
        acc2 = __builtin_amdgcn_wmma_f32_16x16x4_f32(false, a2, false, bones,
                                                     (short)0, acc2, false, false);
    }

    // Every column of D equals the row-sum vector.  Lane n (n<16) holds
    // D[0..7][n]; lane 16+n holds D[8..15][n].  Per-lane 8-sum + xor-16
    // shuffle yields the full plane sum on every lane.
    float s  = acc[0]+acc[1]+acc[2]+acc[3]+acc[4]+acc[5]+acc[6]+acc[7];
    float s2 = acc2[0]+acc2[1]+acc2[2]+acc2[3]+acc2[4]+acc2[5]+acc2[6]+acc2[7];
    s  += __shfl_xor(s,  16, 32);
    s2 += __shfl_xor(s2, 16, 32);

    if (l == 0) {
        const float inv_n = 1.0f / (float)(H2 * W2);
        stats[2 * p + 0] = s * inv_n;            // f2_mean
        stats[2 * p + 1] = s2 - s * s * inv_n;   // sar_ssd = sum((x-mean)^2)
    }
}

// ---------------------------------------------------------------------------
// Kernel B: fused window-sum (via in-LDS integral images) + NCC normalize.
// One workgroup per 64x64 output tile per plane.
// ---------------------------------------------------------------------------
__global__ __launch_bounds__(256)
void ncc_norm_kernel(const float* __restrict__ feat1,
                     const float* __restrict__ xcorr,
                     const float* __restrict__ stats,
                     float* __restrict__ outp)
{
    extern __shared__ float smem[];
    float* Sx  = smem;                    // SATN x SSTRIDE integral of x
    float* Sx2 = smem + SATN * SSTRIDE;   // SATN x SSTRIDE integral of x^2

    const int p   = blockIdx.z;
    const int oi  = blockIdx.y * TILE;
    const int oj  = blockIdx.x * TILE;
    const int fr0 = oi - (H2 - 1);        // feat_1 row of tile's receptive field
    const int fc0 = oj - (W2 - 1);
    const int tid = threadIdx.x;

    const float f2mean = stats[2 * p + 0];
    const float f2ssd  = stats[2 * p + 1];

    const float* f1p = feat1 + (size_t)p * (H1 * W1);
    const float* xcp = xcorr + (size_t)p * (OUTN * OUTN);
    float*       op  = outp  + (size_t)p * (OUTN * OUTN);

    // Prefetch the xcorr tile (gfx1250 global_prefetch_b8) so the streaming
    // read overlaps the SAT construction below.
    {
        int li = tid >> 3;                 // 32 rows covered
        int lj = (tid & 7) * 8;            // 8 points per row -> every line hit
        int i = oi + li, j = oj + lj;
        if (i < OUTN && j < OUTN)
            __builtin_prefetch(&xcp[(size_t)i * OUTN + j], 0, 1);
        i = oi + li + 32;
        if (i < OUTN && j < OUTN)
            __builtin_prefetch(&xcp[(size_t)i * OUTN + j], 0, 1);
    }

    // 1) Load zero-padded 127x127 receptive field into SAT interior.
    for (int idx = tid; idx < RF * RF; idx += 256) {
        int r = idx / RF, c = idx % RF;
        int gr = fr0 + r, gc = fc0 + c;
        float v = 0.0f;
        if ((unsigned)gr < (unsigned)H1 && (unsigned)gc < (unsigned)W1)
            v = f1p[gr * W1 + gc];
        Sx [(r + 1) * SSTRIDE + (c + 1)] = v;
        Sx2[(r + 1) * SSTRIDE + (c + 1)] = v * v;
    }
    // Zero row 0 and column 0.
    for (int idx = tid; idx < SATN; idx += 256) {
        Sx [idx] = 0.0f;              Sx2[idx] = 0.0f;
        Sx [idx * SSTRIDE] = 0.0f;    Sx2[idx * SSTRIDE] = 0.0f;
    }
    __syncthreads();

    // 2) Row-wise inclusive prefix: threads 0..126 -> Sx, 128..254 -> Sx2.
    {
        float* S = (tid < 128) ? Sx : Sx2;
        int r = tid & 127;
        if (r < RF) {
            float run = 0.0f;
            float* row = S + (r + 1) * SSTRIDE;
            for (int c = 1; c <= RF; ++c) { run += row[c]; row[c] = run; }
        }
    }
    __syncthreads();

    // 3) Column-wise inclusive prefix (stride 129 -> conflict-free columns).
    {
        float* S = (tid < 128) ? Sx : Sx2;
        int c = tid & 127;
        if (c < RF) {
            float run = 0.0f;
            float* col = S + (c + 1);
            for (int r = 1; r <= RF; ++r) {
                run += col[r * SSTRIDE];
                col[r * SSTRIDE] = run;
            }
        }
    }
    __syncthreads();

    // 4) Each output pixel = 4 SAT corner lookups per map + pointwise math.
    for (int pidx = tid; pidx < TILE * TILE; pidx += 256) {
        int li = pidx >> 6, lj = pidx & 63;
        int i = oi + li, j = oj + lj;
        if (i >= OUTN || j >= OUTN) continue;

        int rA = li, rB = li + 64;      // window rows [li, li+63] of the tile
        int cA = lj, cB = lj + 64;

        float ws  = Sx [rB * SSTRIDE + cB] - Sx [rA * SSTRIDE + cB]
                  - Sx [rB * SSTRIDE + cA] + Sx [rA * SSTRIDE + cA];
        float ws2 = Sx2[rB * SSTRIDE + cB] - Sx2[rA * SSTRIDE + cB]
                  - Sx2[rB * SSTRIDE + cA] + Sx2[rA * SSTRIDE + cA];

        float xc  = xcp[(size_t)i * OUTN + j];
        float num = xc - ws * f2mean;
        float wc  = ws2 - ws * ws * (1.0f / (float)(H2 * W2));
        float den = __builtin_sqrtf(fmaxf(wc * f2ssd, 0.0f));

        float r = 0.0f;
        if (den > EPSF) {
            float nc = fminf(fmaxf(num, EPSF), FLT_MAX);
            float dc = fminf(den, FLT_MAX);          // den > eps already
            r = nc / dc;
        }
        op[(size_t)i * OUTN + j] = r;
    }
}

// ---------------------------------------------------------------------------
extern "C" void kernel_launch(void* const* d_in, const int* in_sizes, int n_in,
                              void* d_out, int out_size, void* d_ws, size_t ws_size,
                              hipStream_t stream)
{
    const float* feat1 = (const float*)d_in[0];
    const float* feat2 = (const float*)d_in[1];
    const float* xcorr = (const float*)d_in[2];
    float* out   = (float*)d_out;
    float* stats = (float*)d_ws;     // 512 * 2 floats = 4 KB of scratch

    f2_stats_kernel<<<PLANES, 32, 0, stream>>>(feat2, stats);

    dim3 grid((OUTN + TILE - 1) / TILE, (OUTN + TILE - 1) / TILE, PLANES);
    size_t shmem = (size_t)2 * SATN * SSTRIDE * sizeof(float);   // 132 KB
    ncc_norm_kernel<<<grid, 256, shmem, stream>>>(feat1, xcorr, stats, out);
}